// PdhgSolver_54614804136208
// MI455X (gfx1250) — compile-verified
//
#include <hip/hip_runtime.h>

// ---------------------------------------------------------------------------
// PDHG cone-QP solver for MI455X (gfx1250, wave32).
// One wave per environment. Setup matmuls on V_WMMA_F32_16X16X4_F32 with
// zero-padded LDS tiles; Gauss-Jordan solves fully register-resident
// (column-per-lane + v_readlane pivot broadcasts); hot loop register-resident
// with zero LDS traffic (Fm row-per-lane + v_readlane SGPR broadcasts).
// ---------------------------------------------------------------------------

typedef float v2f __attribute__((ext_vector_type(2)));
typedef float v8f __attribute__((ext_vector_type(8)));

#define PDHG_A_MIN (-10.0f)
#define PDHG_A_MAX (10.0f)

// Tile buffers padded to full WMMA extents, odd-dword pitches (conflict-free
// banking); padding zero-filled once so fragment I/O needs no bounds checks.
struct EnvLds {
  float H[32 * 17];   // H (24x12) zero-padded to 32x16, pitch 17
  float G[16 * 17];   // G = H^T H (12x12) in 16x16 tile, pitch 17
  float K[16 * 17];   // (P + G)^-1 zero-padded, pitch 17
  float T[32 * 17];   // H * K (24x12) in 32x16 tiles, pitch 17
  float F[32 * 33];   // Fm = I - T * H^T (24x24) in 32x32 tiles, pitch 33
  float Pi[12 * 25];  // pinv(H) = G^-1 H^T (12x24), pitch 25
  float P[12 * 13];   // P (12x12), pitch 13
};

__device__ __forceinline__ v8f wmma_f32_16x16x4(v2f a, v2f b, v8f c) {
  // (neg_a, A, neg_b, B, c_mod, C, reuse_a, reuse_b)
  return __builtin_amdgcn_wmma_f32_16x16x4_f32(false, a, false, b, (short)0, c,
                                               false, false);
}

// A fragment (16x4 f32): lane L holds A[m][k], m = L%16, k = 2*(L/16)+v.
__device__ __forceinline__ v2f ldA(const float* src, int rs, int cs, int moff,
                                   int kbase, int lane) {
  int m = moff + (lane & 15);
  int k0 = kbase + 2 * (lane >> 4);
  v2f r;
  r.x = src[m * rs + k0 * cs];
  r.y = src[m * rs + (k0 + 1) * cs];
  return r;
}

// B fragment (4x16 f32): lane L holds B[k][n], n = L%16, k = 2*(L/16)+v.
__device__ __forceinline__ v2f ldB(const float* src, int rs, int cs, int kbase,
                                   int noff, int lane) {
  int n = noff + (lane & 15);
  int k0 = kbase + 2 * (lane >> 4);
  v2f r;
  r.x = src[k0 * rs + n * cs];
  r.y = src[(k0 + 1) * rs + n * cs];
  return r;
}

// D tile (16x16 f32): VGPR v holds row (v + 8*(lane/16)), col (lane%16).
__device__ __forceinline__ void stD(float* dst, int ld, int roff, int coff,
                                    v8f d, int lane, bool identity_minus) {
  int c = coff + (lane & 15);
  int rh = roff + 8 * (lane >> 4);
#pragma unroll
  for (int v = 0; v < 8; ++v) {
    int r = rh + v;
    float val = d[v];
    if (identity_minus) val = ((r == c) ? 1.0f : 0.0f) - val;
    dst[r * ld + c] = val;
  }
}

// Uniform-lane broadcast via v_readlane_b32 (SGPR result, no DS traffic).
__device__ __forceinline__ float bcast(float v, int k) {
  union { float f; int i; } u;
  u.f = v;
  union { int i; float f; } w;
  w.i = __builtin_amdgcn_readlane(u.i, k);
  return w.f;
}

// Register-resident Gauss-Jordan on a 12-row augmented system held
// column-per-lane: lane c owns column c in Wc[0..11]; with TAIL, lane c<4
// additionally owns column 32+c in W2[0..11]. Junk columns ride along.
template <bool TAIL>
__device__ __forceinline__ void gj12_reg(float* Wc, float* W2, int lane) {
#pragma unroll
  for (int p = 0; p < 12; ++p) {
    float inv = __builtin_amdgcn_rcpf(bcast(Wc[p], p));
    Wc[p] *= inv;
    if (TAIL) W2[p] *= inv;
#pragma unroll
    for (int r = 0; r < 12; ++r) {
      if (r == p) continue;
      float f = bcast(Wc[r], p);
      Wc[r] = fmaf(-f, Wc[p], Wc[r]);
      if (TAIL) W2[r] = fmaf(-f, W2[p], W2[r]);
    }
  }
}

__global__ __launch_bounds__(128) void pdhg_kernel(
    const float* __restrict__ Pg, const float* __restrict__ qg,
    const float* __restrict__ Hg, const float* __restrict__ bg,
    const float* __restrict__ cfg, const int* __restrict__ itersPtr,
    float* __restrict__ outx, float* __restrict__ outlz, int B) {
  __shared__ EnvLds smem[4];
  const int lane = threadIdx.x & 31;
  const int wid = threadIdx.x >> 5;
  const int env = blockIdx.x * 4 + wid;
  if (env >= B) return;  // wave-uniform

  EnvLds& L = smem[wid];
  const float* Pe = Pg + (size_t)env * 144;
  const int j = lane;
  const int jr = (j < 24) ? j : 23;  // clamped row for inactive lanes
  const int cg = (lane < 12) ? lane : 0;

  // ---- zero padded buffers; load H (24x12) and P (12x12); vectors -------
  for (int i = lane; i < 32 * 17; i += 32) L.H[i] = 0.0f;
  for (int i = lane; i < 16 * 17; i += 32) L.K[i] = 0.0f;
  for (int i = lane; i < 288; i += 32)
    L.H[(i / 12) * 17 + (i % 12)] = Hg[(size_t)env * 288 + i];
  for (int i = lane; i < 144; i += 32)
    L.P[(i / 12) * 13 + (i % 12)] = Pe[i];
  float ql = (j < 12) ? qg[(size_t)env * 12 + j] : 0.0f;
  float bj = (j < 24) ? bg[(size_t)env * 24 + j] : 0.0f;
  float cfj = (j < 24) ? cfg[(size_t)env * 24 + j] : 0.0f;

  // ---- G = H^T H   (M=12, N=12, K=24): 6 WMMAs --------------------------
  {
    v8f acc = {0.f, 0.f, 0.f, 0.f, 0.f, 0.f, 0.f, 0.f};
#pragma unroll
    for (int kb = 0; kb < 24; kb += 4) {
      v2f a = ldA(L.H, /*rs=*/1, /*cs=*/17, 0, kb, lane);  // A = H^T
      v2f b = ldB(L.H, /*rs=*/17, /*cs=*/1, kb, 0, lane);  // B = H
      acc = wmma_f32_16x16x4(a, b, acc);
    }
    stD(L.G, 17, 0, 0, acc, lane, false);
  }

  // ---- K = (P + G)^-1: register GJ on [S | I], column-per-lane ----------
  float Wc[12], W2[12];
  {
#pragma unroll
    for (int r = 0; r < 12; ++r) {
      float sv = L.P[r * 13 + cg] + L.G[r * 17 + cg];
      float iv = (lane - 12 == r) ? 1.0f : 0.0f;
      Wc[r] = (lane < 12) ? sv : iv;
    }
    gj12_reg<false>(Wc, Wc, lane);
    if (lane >= 12 && lane < 24) {
      int c = lane - 12;
#pragma unroll
      for (int r = 0; r < 12; ++r) L.K[r * 17 + c] = Wc[r];
    }
  }

  // ---- T = H * K   (M=24, N=12, K=12): 6 WMMAs --------------------------
#pragma unroll
  for (int mt = 0; mt < 2; ++mt) {
    v8f acc = {0.f, 0.f, 0.f, 0.f, 0.f, 0.f, 0.f, 0.f};
#pragma unroll
    for (int kb = 0; kb < 12; kb += 4) {
      v2f a = ldA(L.H, 17, 1, mt * 16, kb, lane);  // A = H
      v2f b = ldB(L.K, 17, 1, kb, 0, lane);        // B = K
      acc = wmma_f32_16x16x4(a, b, acc);
    }
    stD(L.T, 17, mt * 16, 0, acc, lane, false);
  }

  // ---- Fm = I - T * H^T  (M=24, N=24, K=12): 12 WMMAs -------------------
#pragma unroll
  for (int mt = 0; mt < 2; ++mt) {
#pragma unroll
    for (int nt = 0; nt < 2; ++nt) {
      v8f acc = {0.f, 0.f, 0.f, 0.f, 0.f, 0.f, 0.f, 0.f};
#pragma unroll
      for (int kb = 0; kb < 12; kb += 4) {
        v2f a = ldA(L.T, 17, 1, mt * 16, kb, lane);  // A = T
        v2f b = ldB(L.H, 1, 17, kb, nt * 16, lane);  // B = H^T
        acc = wmma_f32_16x16x4(a, b, acc);
      }
      stD(L.F, 33, mt * 16, nt * 16, acc, lane, true);
    }
  }

  // ---- pq = P^-1 q: register GJ on [P | q] (q in column 12) -------------
  {
#pragma unroll
    for (int r = 0; r < 12; ++r) {
      float pv = L.P[r * 13 + cg];
      float v = (lane < 12) ? pv : 0.0f;
      Wc[r] = (lane == 12) ? bcast(ql, r) : v;
    }
    gj12_reg<false>(Wc, Wc, lane);
  }
  // v = H * pq - b   (pq lives in lane 12's Wc[] -> readlane broadcasts)
  float vj = -bj;
#pragma unroll
  for (int k = 0; k < 12; ++k)
    vj = fmaf(L.H[jr * 17 + k], bcast(Wc[k], 12), vj);

  // ---- Pi = pinv(H) = G^-1 H^T: register GJ on [G | H^T] (36 cols) ------
  {
    int hrow = (lane >= 12) ? (lane - 12) : 0;  // columns 12..31 -> H rows
#pragma unroll
    for (int r = 0; r < 12; ++r) {
      float gv = L.G[r * 17 + cg];
      float hv = L.H[hrow * 17 + r];
      Wc[r] = (lane < 12) ? gv : hv;
      W2[r] = L.H[(20 + (lane & 3)) * 17 + r];  // columns 32..35
    }
    gj12_reg<true>(Wc, W2, lane);
    if (lane >= 12) {  // Pi columns 0..19
      int c = lane - 12;
#pragma unroll
      for (int r = 0; r < 12; ++r) L.Pi[r * 25 + c] = Wc[r];
    }
    if (lane < 4) {  // Pi columns 20..23
#pragma unroll
      for (int r = 0; r < 12; ++r) L.Pi[r * 25 + 20 + lane] = W2[r];
    }
  }

  // ---- Fm row j into registers; mu = Fm * v -----------------------------
  float F[24];
#pragma unroll
  for (int k = 0; k < 24; ++k) F[k] = L.F[jr * 33 + k];
  float muj = 0.0f;
#pragma unroll
  for (int k = 0; k < 24; ++k) muj = fmaf(F[k], bcast(vj, k), muj);

  // ---- PDHG iterations: s = l+z; l' = Fm s + mu; z' = proj(s - 2 l') ----
  // Pure VALU/SALU loop: readlane broadcasts, 4-way split accumulators.
  const int niters = itersPtr[0];
  const bool lt3 = (j < 3), lt6 = (j < 6), lt9 = (j < 9), lt12 = (j < 12);
  float l = 0.0f, z = 0.0f;
  for (int it = 0; it < niters; ++it) {
    float s = l + z;
    float a0 = muj, a1 = 0.0f, a2 = 0.0f, a3 = 0.0f;
#pragma unroll
    for (int k = 0; k < 24; k += 4) {
      a0 = fmaf(F[k + 0], bcast(s, k + 0), a0);
      a1 = fmaf(F[k + 1], bcast(s, k + 1), a1);
      a2 = fmaf(F[k + 2], bcast(s, k + 2), a2);
      a3 = fmaf(F[k + 3], bcast(s, k + 3), a3);
    }
    float lp = (a0 + a1) + (a2 + a3);
    float zp = s - 2.0f * lp;
    // cone z-components for the 4 SOC blocks, selected per lane
    float c0 = bcast(zp, 2), c1 = bcast(zp, 5);
    float c2 = bcast(zp, 8), c3 = bcast(zp, 11);
    float cz = lt6 ? (lt3 ? c0 : c1) : (lt9 ? c2 : c3);
    float zs = (cz > 0.0f) ? zp : 0.0f;                   // SOC mask (j<12)
    float zb = fminf(fmaxf(zp, PDHG_A_MIN), PDHG_A_MAX);  // hardtanh (j>=12)
    z = (lt12 ? zs : zb) * cfj;
    l = lp;
  }

  // ---- x = Pi * (z - b) --------------------------------------------------
  float rres = z - bj;  // lanes >= 24 contribute 0 and are never broadcast
  const int jp = (j < 12) ? j : 0;
  float xj = 0.0f;
#pragma unroll
  for (int k = 0; k < 24; ++k)
    xj = fmaf(L.Pi[jp * 25 + k], bcast(rres, k), xj);

  if (j < 12) outx[(size_t)env * 12 + j] = xj;
  if (j < 24) {
    outlz[(size_t)env * 48 + j] = l;
    outlz[(size_t)env * 48 + 24 + j] = z;
  }
}

extern "C" void kernel_launch(void* const* d_in, const int* in_sizes, int n_in,
                              void* d_out, int out_size, void* d_ws,
                              size_t ws_size, hipStream_t stream) {
  const float* P = (const float*)d_in[0];
  const float* q = (const float*)d_in[1];
  const float* H = (const float*)d_in[2];
  const float* b = (const float*)d_in[3];
  const float* cf = (const float*)d_in[4];
  const int* iters = (const int*)d_in[5];

  const int B = in_sizes[1] / 12;  // q is [B, 12]
  float* outx = (float*)d_out;
  float* outlz = outx + (size_t)B * 12;

  const int blocks = (B + 3) / 4;  // 4 envs (waves) per 128-thread block
  pdhg_kernel<<<blocks, 128, 0, stream>>>(P, q, H, b, cf, iters, outx, outlz,
                                          B);
}